// SSIM_10402410791784
// MI455X (gfx1250) — compile-verified
//
#include <hip/hip_runtime.h>
#include <hip/hip_bf16.h>

// ---------------- SSIM (8x8 box window, VALID) on gfx1250 ----------------
// Memory-bound: 201 MB in -> 1 float out. Tile into LDS (async-to-LDS on
// interior tiles), horizontal box sums in VALU, vertical box sums as a
// banded-ones matmul on V_WMMA_F32_16X16X4_F32, elementwise SSIM evaluated
// directly in the WMMA fragment layout, block reduce + global atomic.
// TH=32 tile halves halo read-amplification vs TH=16 (1.35x vs 1.6x DRAM).

#define WND 8
#define TW 64                 // output cols per tile
#define TH 32                 // output rows per tile
#define IW (TW + WND - 1)     // 71 input cols needed
#define IH (TH + WND - 1)     // 39 input rows needed
#define IN_STRIDE 72          // LDS stride for input tiles (16B aligned)
#define HQ_STRIDE 68          // LDS stride for hsum (bank-conflict-free)
#define IMG 512
#define OUTD 505
#define NTOTF 24482400.0f     // 96 * 505 * 505
#define NTHREADS 256

typedef __attribute__((ext_vector_type(2))) float v2f;
typedef __attribute__((ext_vector_type(8))) float v8f;

#if __has_builtin(__builtin_amdgcn_global_load_async_to_lds_b128)
#define HAVE_ASYNC 1
#else
#define HAVE_ASYNC 0
#endif

// Parameter types per hipcc diagnostic: pointer to GCC-vector of 4 ints,
// AS1 (global) source, AS3 (LDS) destination.
typedef int v4i __attribute__((vector_size(4 * sizeof(int))));
typedef __attribute__((address_space(1))) v4i GV4;
typedef __attribute__((address_space(3))) v4i LV4;

__device__ __forceinline__ void wait_async0() {
#if __has_builtin(__builtin_amdgcn_s_wait_asynccnt)
  __builtin_amdgcn_s_wait_asynccnt(0);
#else
  asm volatile("s_wait_asynccnt 0" ::: "memory");
#endif
}

__launch_bounds__(NTHREADS)
__global__ void ssim_main(const float* __restrict__ gt,
                          const float* __restrict__ ny,
                          float* __restrict__ acc) {
  __shared__ float s_gt[IH * IN_STRIDE];             // 11.2 KB
  __shared__ float s_ny[IH * IN_STRIDE];             // 11.2 KB
  __shared__ float s_h[5][IH * HQ_STRIDE];           // 53.0 KB
  __shared__ float s_red[NTHREADS];

  const int tid  = threadIdx.x;
  const int tx   = blockIdx.x;   // 0..7
  const int ty   = blockIdx.y;   // 0..15
  const int img  = blockIdx.z;   // 0..95
  const int row0 = ty * TH;
  const int col0 = tx * TW;
  const float* gbase = gt + (size_t)img * IMG * IMG;
  const float* nbase = ny + (size_t)img * IMG * IMG;

  // ---- Phase 1: stage input tiles into LDS ----
  const bool interior = (row0 + IH <= IMG) && (col0 + IN_STRIDE <= IMG);
#if HAVE_ASYNC
  if (interior) {
    const int chunksPerRow = IN_STRIDE / 4;       // 18 x b128 per row
    const int total = IH * chunksPerRow;          // 702
    for (int i = tid; i < total; i += NTHREADS) {
      int r  = i / chunksPerRow;
      int c4 = (i % chunksPerRow) * 4;
      const float* g = gbase + (size_t)(row0 + r) * IMG + col0 + c4;
      const float* n = nbase + (size_t)(row0 + r) * IMG + col0 + c4;
      __builtin_amdgcn_global_load_async_to_lds_b128(
          (GV4*)g, (LV4*)&s_gt[r * IN_STRIDE + c4], 0, 0);
      __builtin_amdgcn_global_load_async_to_lds_b128(
          (GV4*)n, (LV4*)&s_ny[r * IN_STRIDE + c4], 0, 0);
    }
    wait_async0();
  } else
#endif
  {
    for (int i = tid; i < IH * IN_STRIDE; i += NTHREADS) {
      int r = i / IN_STRIDE, c = i % IN_STRIDE;
      int grow = min(row0 + r, IMG - 1);
      int gcol = min(col0 + c, IMG - 1);
      s_gt[i] = gbase[(size_t)grow * IMG + gcol];
      s_ny[i] = nbase[(size_t)grow * IMG + gcol];
    }
  }
  __syncthreads();

  // ---- Phase 2: horizontal 8-wide box sums of 5 quantities ----
  for (int i = tid; i < IH * TW; i += NTHREADS) {
    int r = i / TW, c = i % TW;
    const float* gp = &s_gt[r * IN_STRIDE + c];
    const float* np = &s_ny[r * IN_STRIDE + c];
    float sx = 0.f, sy = 0.f, sxx = 0.f, syy = 0.f, sxy = 0.f;
#pragma unroll
    for (int k = 0; k < WND; ++k) {
      float x = gp[k], y = np[k];
      sx += x; sy += y; sxx += x * x; syy += y * y; sxy += x * y;
    }
    int o = r * HQ_STRIDE + c;
    s_h[0][o] = sx; s_h[1][o] = sy; s_h[2][o] = sxx; s_h[3][o] = syy; s_h[4][o] = sxy;
  }
  __syncthreads();

  // ---- Phase 3: vertical box sums as banded matmul on WMMA f32 16x16x4 ----
  // D(16x16) = A_band(16x24) x H(24x16), A[m][k] = (k>=m && k<m+8) ? 1 : 0.
  // 8 waves cover 2 (rows) x 4 (cols) 16x16 output subtiles.
  const int wave = tid >> 5;         // 0..7
  const int lane = tid & 31;
  const int hi   = lane >> 4;        // lane half
  const int m    = lane & 15;        // A-row / also B-col index
  const int sm   = (wave >> 2) * 16; // subtile row offset (0 or 16)
  const int sn   = (wave & 3) * 16;  // subtile col offset

  v8f accX = {}, accY = {}, accXX = {}, accYY = {}, accXY = {};
#pragma unroll
  for (int kc = 0; kc < 6; ++kc) {
    const int k0r = kc * 4 + 2 * hi;         // K index held in VGPR0 by this lane
    const int k1r = k0r + 1;                 // K index held in VGPR1
    const float a0 = (k0r >= m && k0r < m + WND) ? 1.0f : 0.0f;
    const float a1 = (k1r >= m && k1r < m + WND) ? 1.0f : 0.0f;
    const int r0 = sm + k0r;                     // hsum row (<= 38)
    const int r1 = (sm + k1r < IH) ? (sm + k1r) : (IH - 1); // k=23 pad: a1==0
    const v2f A = {a0, a1};
#define DO_Q(ACC, Q)                                                        \
    {                                                                       \
      v2f B = { s_h[Q][r0 * HQ_STRIDE + sn + m],                            \
                s_h[Q][r1 * HQ_STRIDE + sn + m] };                          \
      ACC = __builtin_amdgcn_wmma_f32_16x16x4_f32(                          \
          false, A, false, B, (short)0, ACC, false, false);                 \
    }
    DO_Q(accX, 0) DO_Q(accY, 1) DO_Q(accXX, 2) DO_Q(accYY, 3) DO_Q(accXY, 4)
#undef DO_Q
  }

  // ---- Elementwise SSIM directly in fragment layout ----
  const float C1v = 1e-4f;   // (0.01*1.0)^2
  const float C2v = 9e-4f;   // (0.03*1.0)^2
  float part = 0.0f;
#pragma unroll
  for (int v = 0; v < 8; ++v) {
    int orow = row0 + sm + v + 8 * hi; // global output row of this fragment elt
    int ocol = col0 + sn + m;          // global output col
    if (orow < OUTD && ocol < OUTD) {
      float mu1 = accX[v], mu2 = accY[v];
      float mu1sq = mu1 * mu1, mu2sq = mu2 * mu2, mu12 = mu1 * mu2;
      float s1  = accXX[v] - mu1sq;
      float s2  = accYY[v] - mu2sq;
      float s12 = accXY[v] - mu12;
      float num = (2.0f * mu12 + C1v) * (2.0f * s12 + C2v);
      float den = (mu1sq + mu2sq + C1v) * (s1 + s2 + C2v);
      part += num / den;
    }
  }

  // ---- Block reduce + one atomic per workgroup ----
  s_red[tid] = part;
  __syncthreads();
  for (int s = NTHREADS / 2; s > 0; s >>= 1) {
    if (tid < s) s_red[tid] += s_red[tid + s];
    __syncthreads();
  }
  if (tid == 0) atomicAdd(acc, s_red[0]);
}

__global__ void ssim_init(float* acc) { acc[0] = 0.0f; }

__global__ void ssim_fin(const float* __restrict__ acc, float* __restrict__ out) {
  out[0] = 1.0f - acc[0] / NTOTF;
}

extern "C" void kernel_launch(void* const* d_in, const int* in_sizes, int n_in,
                              void* d_out, int out_size, void* d_ws, size_t ws_size,
                              hipStream_t stream) {
  const float* gt = (const float*)d_in[0];
  const float* ny = (const float*)d_in[1];
  float* out = (float*)d_out;
  float* acc = (float*)d_ws;

  ssim_init<<<1, 1, 0, stream>>>(acc);
  dim3 grid(8, 16, 96);   // tx tiles (505/64), ty tiles (505/32), 32*3 images
  ssim_main<<<grid, NTHREADS, 0, stream>>>(gt, ny, acc);
  ssim_fin<<<1, 1, 0, stream>>>(acc, out);
}